// Encoder_31069793419699
// MI455X (gfx1250) — compile-verified
//
#include <hip/hip_runtime.h>
#include <hip/hip_bf16.h>

typedef __attribute__((ext_vector_type(2))) float v2f;
typedef __attribute__((ext_vector_type(8))) float v8f;

#define D 128
#define WSTRIDE 132   // LDS row stride for transposed W: %64 banks == 4 -> conflict-free b64 loads

// ---------------- degree / normalization ----------------

__global__ void deg_init(float* deg, int n) {
    int i = blockIdx.x * blockDim.x + threadIdx.x;
    if (i < n) deg[i] = 1.0f;            // self-loop weight
}

__global__ void deg_accum(const int* __restrict__ dst, const float* __restrict__ w,
                          float* deg, int e) {
    int i = blockIdx.x * blockDim.x + threadIdx.x;
    if (i < e) atomicAdd(&deg[dst[i]], w[i]);
}

__global__ void deg_finalize(float* deg, int n) {   // deg -> dinv in place
    int i = blockIdx.x * blockDim.x + threadIdx.x;
    if (i < n) {
        float d = deg[i];
        deg[i] = (d > 0.0f) ? rsqrtf(d) : 0.0f;
    }
}

// ---------------- CSR-by-destination construction (built once, reused) ----------------

__global__ void zero_int(int* p, int n) {
    int i = blockIdx.x * blockDim.x + threadIdx.x;
    if (i < n) p[i] = 0;
}

__global__ void hist_count(const int* __restrict__ dst, int* __restrict__ cnt, int e) {
    int i = blockIdx.x * blockDim.x + threadIdx.x;
    if (i < e) atomicAdd(&cnt[dst[i]], 1);
}

// S0: per-block (512 elements) sums
__global__ __launch_bounds__(512) void block_sums(const int* __restrict__ cnt,
                                                  int* __restrict__ bsum, int n) {
    __shared__ int s[512];
    int t = threadIdx.x;
    int g = blockIdx.x * 512 + t;
    s[t] = (g < n) ? cnt[g] : 0;
    __syncthreads();
    for (int off = 256; off > 0; off >>= 1) {
        if (t < off) s[t] += s[t + off];
        __syncthreads();
    }
    if (t == 0) bsum[blockIdx.x] = s[0];
}

// S1: serial exclusive scan of block sums (nb ~ 196) + write rowptr[n] = total
__global__ void serial_scan(int* bsum, int nb, int* rowptr, int n) {
    int run = 0;
    for (int b = 0; b < nb; ++b) {
        int v = bsum[b];
        bsum[b] = run;
        run += v;
    }
    rowptr[n] = run;
}

// S2: per-block inclusive scan -> global exclusive offsets
__global__ __launch_bounds__(512) void block_scan(const int* __restrict__ cnt,
                                                  const int* __restrict__ bsumExcl,
                                                  int* __restrict__ rowptr, int n) {
    __shared__ int s[512];
    int t = threadIdx.x;
    int g = blockIdx.x * 512 + t;
    int v = (g < n) ? cnt[g] : 0;
    s[t] = v;
    __syncthreads();
    for (int off = 1; off < 512; off <<= 1) {
        int u = (t >= off) ? s[t - off] : 0;
        __syncthreads();
        s[t] += u;
        __syncthreads();
    }
    if (g < n) rowptr[g] = bsumExcl[blockIdx.x] + s[t] - v;   // exclusive
}

// Fill packed (src, coef) stream ordered by destination. cur[] pre-zeroed.
__global__ void csr_fill(const int* __restrict__ src, const int* __restrict__ dst,
                         const float* __restrict__ w, const float* __restrict__ dinv,
                         const int* __restrict__ rowptr, int* __restrict__ cur,
                         int2* __restrict__ edata, int e) {
    int i = blockIdx.x * blockDim.x + threadIdx.x;
    if (i >= e) return;
    int s = src[i];
    int d = dst[i];
    int pos = atomicAdd(&cur[d], 1);
    int idx = rowptr[d] + pos;
    float c = dinv[s] * w[i] * dinv[d];
    edata[idx] = make_int2(s, __float_as_int(c));
}

// ---------------- GEMM: H[N,128] = X[N,128] @ W[128,128] via fp32 WMMA ----------------
// Block = 256 threads = 8 waves; block covers 128 rows; wave computes 16 rows x 128 cols.
// W transposed in LDS (stride 132): B operand = one aligned ds_load_b64, bank-conflict-free.

__global__ __launch_bounds__(256) void gemm_wmma(const float* __restrict__ X,
                                                 const float* __restrict__ W,
                                                 float* __restrict__ H, int nrows) {
    __shared__ float sW[D * WSTRIDE];    // 66 KB, transposed: sW[col*132 + k] = W[k*128 + col]
    const int tid = threadIdx.x;
    for (int i = tid; i < D * D; i += 256) {
        int k = i >> 7, col = i & 127;
        sW[col * WSTRIDE + k] = W[i];
    }
    __syncthreads();

    const int wave = tid >> 5;
    const int lane = tid & 31;
    const int row0 = (blockIdx.x * 8 + wave) * 16;

    // A 16x4 fp32: lanes 0-15 M=lane K={0,1}; lanes 16-31 M=lane-16 K={2,3}
    const int m  = lane & 15;
    const int kb = (lane >> 4) * 2;      // 0 or 2
    int r = row0 + m;
    if (r >= nrows) r = nrows - 1;       // clamp loads; stores guarded
    const float* __restrict__ xrow = X + (size_t)r * D;

    // B 4x16 fp32: lanes 0-15 N=lane K={0,1}; lanes 16-31 N=lane-16 K={2,3}
    const int n = lane & 15;

    v8f acc[8] = {};
    for (int k = 0; k < D; k += 4) {
        v2f a;
        a.x = xrow[k + kb];
        a.y = xrow[k + kb + 1];
#pragma unroll
        for (int c = 0; c < 8; ++c) {
            // contiguous (k+kb, k+kb+1) pair, 8B aligned -> ds_load_b64
            v2f b = *reinterpret_cast<const v2f*>(&sW[(c * 16 + n) * WSTRIDE + k + kb]);
            acc[c] = __builtin_amdgcn_wmma_f32_16x16x4_f32(
                false, a, false, b, (short)0, acc[c], false, false);
        }
    }

    if (row0 < nrows) {
        const int mhalf = (lane >> 4) * 8;
#pragma unroll
        for (int r8 = 0; r8 < 8; ++r8) {
            int row = row0 + r8 + mhalf;
            if (row < nrows) {
                float* __restrict__ hrow = H + (size_t)row * D + n;
#pragma unroll
                for (int c = 0; c < 8; ++c) hrow[c * 16] = acc[c][r8];
            }
        }
    }
}

// ---------------- aggregation: OUT[d] = dinv[d]^2*H[d] + sum_e coef_e * H[src_e] ------
// One wave per node (node = blockIdx.x -> wave-uniform edge stream -> scalar loads).
// Lane owns 4 contiguous floats; accumulate in registers; single store. No atomics.

__global__ __launch_bounds__(32) void aggregate(const float* __restrict__ H,
                                                const float* __restrict__ dinv,
                                                const int* __restrict__ rowptr,
                                                const int2* __restrict__ edata,
                                                float* __restrict__ OUT) {
    const int node = blockIdx.x;
    const int lane = threadIdx.x;
    const float di = dinv[node];
    const float self = di * di;
    float4 acc = ((const float4*)(H + (size_t)node * D))[lane];
    acc.x *= self; acc.y *= self; acc.z *= self; acc.w *= self;

    const int beg = rowptr[node];
    const int end = rowptr[node + 1];
    for (int j = beg; j < end; ++j) {
        int2 p = edata[j];                    // (src, coef) in one 8B load
        float c = __int_as_float(p.y);
        float4 v = ((const float4*)(H + (size_t)p.x * D))[lane];
        acc.x += c * v.x; acc.y += c * v.y; acc.z += c * v.z; acc.w += c * v.w;
    }
    ((float4*)(OUT + (size_t)node * D))[lane] = acc;
}

// ---------------- batch-norm ----------------

__global__ void zero_stats(float* stats) {
    stats[threadIdx.x] = 0.0f;           // 256 threads: sum[128] + sumsq[128]
}

__global__ __launch_bounds__(128) void col_stats(const float* __restrict__ X,
                                                 float* __restrict__ stats, int n) {
    const int j = threadIdx.x;           // coalesced: lanes read consecutive columns
    float s = 0.0f, ss = 0.0f;
    for (int r = blockIdx.x; r < n; r += gridDim.x) {
        float v = X[(size_t)r * D + j];
        s += v; ss += v * v;
    }
    atomicAdd(&stats[j], s);
    atomicAdd(&stats[D + j], ss);
}

// bias b cancels inside BN exactly: BN(y+b) == BN(y). Fold gamma/beta into scale/shift.
__global__ void bn_finalize(const float* __restrict__ stats, const float* __restrict__ gamma,
                            const float* __restrict__ beta, float* __restrict__ ss, int n) {
    const int j = threadIdx.x;           // 128 threads
    float invn = 1.0f / (float)n;
    float mean = stats[j] * invn;
    float var  = stats[D + j] * invn - mean * mean;
    float sc = gamma[j] * rsqrtf(var + 1e-5f);
    ss[j] = sc;
    ss[D + j] = beta[j] - mean * sc;
}

__global__ void bn_relu_apply(const float* __restrict__ X, const float* __restrict__ ss,
                              float* __restrict__ Y, int n) {
    size_t i = (size_t)blockIdx.x * blockDim.x + threadIdx.x;   // over N*32 float4s
    if (i >= (size_t)n * 32) return;
    int j4 = (int)(i & 31) * 4;
    float4 v = ((const float4*)X)[i];
    v.x = fmaxf(v.x * ss[j4 + 0] + ss[D + j4 + 0], 0.0f);
    v.y = fmaxf(v.y * ss[j4 + 1] + ss[D + j4 + 1], 0.0f);
    v.z = fmaxf(v.z * ss[j4 + 2] + ss[D + j4 + 2], 0.0f);
    v.w = fmaxf(v.w * ss[j4 + 3] + ss[D + j4 + 3], 0.0f);
    ((float4*)Y)[i] = v;
}

// ---------------- driver ----------------

extern "C" void kernel_launch(void* const* d_in, const int* in_sizes, int n_in,
                              void* d_out, int out_size, void* d_ws, size_t ws_size,
                              hipStream_t stream) {
    const float* x     = (const float*)d_in[0];
    const int*   ei    = (const int*)d_in[1];
    const float* ew    = (const float*)d_in[2];
    const float* W1    = (const float*)d_in[3];
    // d_in[4] = b1 (cancels in BN)
    const float* W2    = (const float*)d_in[5];
    // d_in[6] = b2 (cancels in BN)
    const float* g1    = (const float*)d_in[7];
    const float* beta1 = (const float*)d_in[8];
    const float* g2    = (const float*)d_in[9];
    const float* beta2 = (const float*)d_in[10];
    float* out = (float*)d_out;

    const int N = in_sizes[0] / D;
    const int E = in_sizes[2];
    const int* src = ei;
    const int* dst = ei + E;

    // workspace layout
    float* A      = (float*)d_ws;                    // N*D
    float* B      = A + (size_t)N * D;               // N*D
    float* dinv   = B + (size_t)N * D;               // N
    float* stats  = dinv + N;                        // 256
    float* sscal  = stats + 256;                     // 256
    int*   rowptr = (int*)(sscal + 256);             // N+1
    int*   cnt    = rowptr + (N + 1);                // N (then reused as cursor)
    int*   bsum   = cnt + N;                         // nb
    int2*  edata  = (int2*)(bsum + 512);             // E pairs (src, coef)

    const int nb     = (N + 511) / 512;
    const int nvec   = N * 32;
    const int gVec   = (nvec + 255) / 256;
    const int gNode  = (N + 255) / 256;
    const int gEdge  = (E + 255) / 256;
    const int gGemm  = (N + 127) / 128;

    // ---- normalization (shared by both layers) ----
    deg_init<<<gNode, 256, 0, stream>>>(dinv, N);
    deg_accum<<<gEdge, 256, 0, stream>>>(dst, ew, dinv, E);
    deg_finalize<<<gNode, 256, 0, stream>>>(dinv, N);

    // ---- CSR by destination (built once) ----
    zero_int<<<gNode, 256, 0, stream>>>(cnt, N);
    hist_count<<<gEdge, 256, 0, stream>>>(dst, cnt, E);
    block_sums<<<nb, 512, 0, stream>>>(cnt, bsum, N);
    serial_scan<<<1, 1, 0, stream>>>(bsum, nb, rowptr, N);
    block_scan<<<nb, 512, 0, stream>>>(cnt, bsum, rowptr, N);
    zero_int<<<gNode, 256, 0, stream>>>(cnt, N);     // reuse as per-node cursor
    csr_fill<<<gEdge, 256, 0, stream>>>(src, dst, ew, dinv, rowptr, cnt, edata, E);

    // ---- layer 1 ----
    gemm_wmma<<<gGemm, 256, 0, stream>>>(x, W1, A, N);                  // A = x@W1
    aggregate<<<N, 32, 0, stream>>>(A, dinv, rowptr, edata, B);         // B = norm-agg(A)
    zero_stats<<<1, 256, 0, stream>>>(stats);
    col_stats<<<1024, 128, 0, stream>>>(B, stats, N);
    bn_finalize<<<1, 128, 0, stream>>>(stats, g1, beta1, sscal, N);
    bn_relu_apply<<<gVec, 256, 0, stream>>>(B, sscal, A, N);            // A = y1

    // ---- layer 2 ----
    gemm_wmma<<<gGemm, 256, 0, stream>>>(A, W2, B, N);                  // B = y1@W2
    aggregate<<<N, 32, 0, stream>>>(B, dinv, rowptr, edata, A);         // A = norm-agg(B)
    zero_stats<<<1, 256, 0, stream>>>(stats);
    col_stats<<<1024, 128, 0, stream>>>(A, stats, N);
    bn_finalize<<<1, 128, 0, stream>>>(stats, g2, beta2, sscal, N);
    bn_relu_apply<<<gVec, 256, 0, stream>>>(A, sscal, out, N);
}